// Loss_16217796510388
// MI455X (gfx1250) — compile-verified
//
#include <hip/hip_runtime.h>
#include <math.h>

// ---------------------------------------------------------------------------
// Loss kernel for MI455X (gfx1250, wave32).
// Bandwidth-bound: 5 x 64MiB f32 streams read exactly once (~335MB -> ~14us
// at 23.3 TB/s). Single-pass per array; deterministic tree reductions.
// CDNA5 paths used:
//   * V_WMMA_F32_16X16X4_F32 as a 64-way cross-lane adder (B = ones).
//   * global_load_async_to_lds_b128 + s_wait_asynccnt row staging (ASYNCcnt).
// ---------------------------------------------------------------------------

#define C_CH 32768
#define L_LEN 512
#define N_TOT (C_CH * L_LEN) /* 16777216 */
#define KELV 273.15f

typedef __attribute__((ext_vector_type(2))) float v2f;
typedef __attribute__((ext_vector_type(8))) float v8f;
typedef int i4 __attribute__((vector_size(16)));         // int4 as the builtin wants
typedef __attribute__((address_space(1))) i4 gi4;        // global int4
typedef __attribute__((address_space(3))) i4 si4;        // LDS int4

#if defined(__HIP_DEVICE_COMPILE__) && __has_builtin(__builtin_amdgcn_wmma_f32_16x16x4_f32)
#define HAVE_WMMA_F32X4 1
#else
#define HAVE_WMMA_F32X4 0
#endif

#if defined(__HIP_DEVICE_COMPILE__) && \
    __has_builtin(__builtin_amdgcn_global_load_async_to_lds_b128) && \
    __has_builtin(__builtin_amdgcn_s_wait_asynccnt)
#define HAVE_ASYNC_LDS 1
#else
#define HAVE_ASYNC_LDS 0
#endif

__device__ __forceinline__ float relu(float x) { return fmaxf(x, 0.0f); }

// Sum of (x,y) over all 32 lanes, broadcast to every lane.
// WMMA 16x16x4 f32 with B = ones: D[m][n] = x_m + y_m + x_{m+16} + y_{m+16}.
// Lane t<16 holds D rows 0..7 (8 accumulator VGPRs), lane t>=16 rows 8..15;
// summing the 8 D components + one shfl_xor(16) yields the full 64-way sum.
// Must be called with EXEC all-ones (wave-uniform control flow).
__device__ __forceinline__ float wave_sum2(float x, float y) {
#if HAVE_WMMA_F32X4
  v2f a; a[0] = x;    a[1] = y;
  v2f b; b[0] = 1.0f; b[1] = 1.0f;
  v8f c = {};
  c = __builtin_amdgcn_wmma_f32_16x16x4_f32(false, a, false, b,
                                            (short)0, c, false, false);
  float s = ((c[0] + c[1]) + (c[2] + c[3])) + ((c[4] + c[5]) + (c[6] + c[7]));
  s += __shfl_xor(s, 16, 32);
  return s;
#else
  float s = x + y;
  #pragma unroll
  for (int off = 16; off > 0; off >>= 1) s += __shfl_xor(s, off, 32);
  return s;
#endif
}

// 256-thread block sum (8 waves); deterministic fixed-order combine.
__device__ __forceinline__ float block_sum2(float x, float y, float* sc) {
  const int lane = threadIdx.x & 31;
  const int wv   = threadIdx.x >> 5;
  const float wsum = wave_sum2(x, y);
  __syncthreads();
  if (lane == 0) sc[wv] = wsum;
  __syncthreads();
  float tot = 0.0f;
  #pragma unroll
  for (int i = 0; i < 8; ++i) tot += sc[i];
  return tot;
}

// --------------------------- sum((An - Ar)^2) ------------------------------
__global__ __launch_bounds__(256) void k_mse(const float* __restrict__ An,
                                             const float* __restrict__ Ar,
                                             float* __restrict__ part) {
  __shared__ float sc[8];
  const int tid = blockIdx.x * 256 + threadIdx.x; // 0..262143, exact coverage
  float a0 = 0.0f, a1 = 0.0f;
  #pragma unroll
  for (int i = 0; i < 16; ++i) {
    const int q = (tid + i * 262144) * 4;
    const float4 x = *reinterpret_cast<const float4*>(An + q);
    const float4 r = *reinterpret_cast<const float4*>(Ar + q);
    const float d0 = x.x - r.x, d1 = x.y - r.y;
    const float d2 = x.z - r.z, d3 = x.w - r.w;
    const float s = (d0 * d0 + d1 * d1) + (d2 * d2 + d3 * d3);
    if (i & 1) a1 += s; else a0 += s;
  }
  const float tot = block_sum2(a0, a1, sc);
  if (threadIdx.x == 0) part[blockIdx.x] = tot;
}

// --------------------------- per-row penalties -----------------------------
// One wave32 per row; 8 rows per block. Lane l owns cols [16l, 16l+16).
__global__ __launch_bounds__(256) void k_row(const float* __restrict__ Aj,
                                             const float* __restrict__ Ac,
                                             const float* __restrict__ Ap,
                                             const unsigned char* __restrict__ mFitAp,
                                             const unsigned char* __restrict__ mFitAj,
                                             float* __restrict__ part) {
#if HAVE_ASYNC_LDS
  __shared__ __align__(16) float sAj[8][L_LEN];
  __shared__ __align__(16) float sAc[8][L_LEN];
  __shared__ __align__(16) float sAp[8][L_LEN];
#endif
  __shared__ float sterm[8];
  const int lane = threadIdx.x & 31;
  const int wv   = threadIdx.x >> 5;
  const int row  = blockIdx.x * 8 + wv;
  const int base = row * L_LEN;

  const float *pj, *pc, *pp;
#if HAVE_ASYNC_LDS
  // Stage the three row tiles Global->LDS via ASYNCcnt-tracked copies
  // (bypasses VGPR load-return; per-wave wait, no cross-wave barrier needed).
  #pragma unroll
  for (int k = 0; k < 4; ++k) {
    const int col = lane * 16 + k * 4;
    __builtin_amdgcn_global_load_async_to_lds_b128(
        (gi4*)(Aj + base + col), (si4*)(&sAj[wv][col]), 0, 0);
    __builtin_amdgcn_global_load_async_to_lds_b128(
        (gi4*)(Ac + base + col), (si4*)(&sAc[wv][col]), 0, 0);
    __builtin_amdgcn_global_load_async_to_lds_b128(
        (gi4*)(Ap + base + col), (si4*)(&sAp[wv][col]), 0, 0);
  }
  __builtin_amdgcn_s_wait_asynccnt(0);
  pj = &sAj[wv][0]; pc = &sAc[wv][0]; pp = &sAp[wv][0];
#else
  pj = Aj + base; pc = Ac + base; pp = Ap + base;
#endif

  float apn0 = 0.f, apn1 = 0.f, dn0 = 0.f, dn1 = 0.f, dp0 = 0.f, dp1 = 0.f;
  float bestV = 3.402823466e38f;
  int   bestI = 0x7fffffff;
  #pragma unroll
  for (int k = 0; k < 4; ++k) {
    const int col = lane * 16 + k * 4;
    const float4 aj = *reinterpret_cast<const float4*>(pj + col);
    const float4 ac = *reinterpret_cast<const float4*>(pc + col);
    const float4 ap = *reinterpret_cast<const float4*>(pp + col);
    const float ajv[4] = {aj.x, aj.y, aj.z, aj.w};
    const float acv[4] = {ac.x, ac.y, ac.z, ac.w};
    const float apv[4] = {ap.x, ap.y, ap.z, ap.w};
    float apn = 0.f, dn = 0.f, dp = 0.f;
    #pragma unroll
    for (int j = 0; j < 4; ++j) {
      const int ccol = col + j;
      apn += relu(-apv[j]);                    // relu(-Ap) over all 512 cols
      const float d  = acv[j] - ajv[j];        // diff only over cols 0..510
      const float in = (ccol < (L_LEN - 1)) ? 1.0f : 0.0f;
      dn += relu(-d) * in;
      dp += relu(d) * in;
      const float ad = fabsf(d);
      if ((ccol < (L_LEN - 1)) && (ad < bestV)) { bestV = ad; bestI = ccol; }
    }
    if (k & 1) { apn1 += apn; dn1 += dn; dp1 += dp; }
    else       { apn0 += apn; dn0 += dn; dp0 += dp; }
  }

  // Wave-level reductions (WMMA adder); argmin via shuffle with
  // first-occurrence tie-break to match jnp.argmin.
  const float sApNeg = wave_sum2(apn0, apn1);
  const float lsAj   = wave_sum2(dn0, dn1);
  const float lsAc   = wave_sum2(dp0, dp1);
  #pragma unroll
  for (int off = 16; off > 0; off >>= 1) {
    const float ov = __shfl_xor(bestV, off, 32);
    const int   oi = __shfl_xor(bestI, off, 32);
    if (ov < bestV || (ov == bestV && oi < bestI)) { bestV = ov; bestI = oi; }
  }

  if (lane == 0) {
    const float ajEnd = pj[L_LEN - 1];
    const float acEnd = pc[L_LEN - 1];
    const float apEnd = pp[L_LEN - 1];
    const float ajInt = pj[bestI];
    const float apInt = pp[bestI];
    const float mAp = mFitAp[row] ? 1.0f : 0.0f;
    const float mAj = mFitAj[row] ? 1.0f : 0.0f;
    float term = 5.0f * relu(ajInt * 1.1f - apInt);
    term += (relu(8.0f - lsAj) + relu(8.0f - lsAc)) * mAj;
    term += relu(apEnd - ajEnd) * mAp * 0.15f;
    term += relu(ajEnd - acEnd) * mAj;
    term += sApNeg;
    sterm[wv] = term;
  }
  __syncthreads();
  if (threadIdx.x == 0) {
    float t = 0.0f;
    #pragma unroll
    for (int i = 0; i < 8; ++i) t += sterm[i];
    part[blockIdx.x] = t;
  }
}

// ------------------- masked raw moments for correlations -------------------
__global__ __launch_bounds__(256) void k_mom(const float* __restrict__ Vc,
                                             const float* __restrict__ Jm,
                                             const float* __restrict__ Rd,
                                             const unsigned char* __restrict__ mNl,
                                             float* __restrict__ part) {
  __shared__ float sm[8][10];
  const int t = blockIdx.x * 256 + threadIdx.x; // 0..8191
  float acc[10][2];
  #pragma unroll
  for (int q = 0; q < 10; ++q) { acc[q][0] = 0.f; acc[q][1] = 0.f; }
  #pragma unroll
  for (int i = 0; i < 4; ++i) {
    const int e = t + i * 8192;
    const float m = mNl[e] ? 1.0f : 0.0f;
    const float v = Vc[e], j = Jm[e], r = Rd[e];
    const int s = i >> 1;
    acc[0][s] += m;
    acc[1][s] += v * m;
    acc[2][s] += j * m;
    acc[3][s] += r * m;
    acc[4][s] += v * v * m;
    acc[5][s] += j * j * m;
    acc[6][s] += r * r * m;
    acc[7][s] += v * j * m;
    acc[8][s] += v * r * m;
    acc[9][s] += relu(-r);
  }
  const int lane = threadIdx.x & 31, wv = threadIdx.x >> 5;
  #pragma unroll
  for (int q = 0; q < 10; ++q) {
    const float wsum = wave_sum2(acc[q][0], acc[q][1]);
    if (lane == 0) sm[wv][q] = wsum;
  }
  __syncthreads();
  if (threadIdx.x < 10) {
    float s = 0.0f;
    #pragma unroll
    for (int w = 0; w < 8; ++w) s += sm[w][threadIdx.x];
    part[blockIdx.x * 10 + threadIdx.x] = s;
  }
}

// ------------------------------- finalize ----------------------------------
__global__ __launch_bounds__(256) void k_final(const float* __restrict__ pMse,
                                               const float* __restrict__ pRow,
                                               const float* __restrict__ pMom,
                                               const float* __restrict__ dHaV,
                                               const float* __restrict__ dHaJ,
                                               const float* __restrict__ dHaT,
                                               const float* __restrict__ ToV,
                                               const float* __restrict__ ToJ,
                                               const float* __restrict__ ToT,
                                               const float* __restrict__ gm,
                                               float* __restrict__ out) {
  __shared__ float sc[8];
  const int t = threadIdx.x;
  float a0 = 0.f, a1 = 0.f;
  #pragma unroll
  for (int i = 0; i < 4; ++i) {
    const float v = pMse[t + i * 256];
    if (i & 1) a1 += v; else a0 += v;
  }
  const float sse = block_sum2(a0, a1, sc);
  float r0 = 0.f, r1 = 0.f;
  #pragma unroll
  for (int i = 0; i < 16; ++i) {
    const float v = pRow[t + i * 256];
    if (i & 1) r1 += v; else r0 += v;
  }
  const float rowSum = block_sum2(r0, r1, sc);

  if (t == 0) {
    float M[10];
    for (int q = 0; q < 10; ++q) {
      float s = 0.0f;
      for (int b = 0; b < 32; ++b) s += pMom[b * 10 + q];
      M[q] = s;
    }
    const float n = M[0];
    float loss = sse * (10.0f / (float)N_TOT);
    { // corr(Jmax25, Vcmax25), target 0.7
      const float cxy = M[7] - M[2] * M[1] / n;
      const float cx  = M[5] - M[2] * M[2] / n;
      const float cy  = M[4] - M[1] * M[1] / n;
      float cost = cxy / (sqrtf(cx) * sqrtf(cy));
      if (cost != cost) cost = 0.0f; // NaN -> 0
      loss += 0.7f - fminf(cost, 0.7f);
    }
    { // corr(Rd25, Vcmax25), target 0.4
      const float cxy = M[8] - M[3] * M[1] / n;
      const float cx  = M[6] - M[3] * M[3] / n;
      const float cy  = M[4] - M[1] * M[1] / n;
      float cost = cxy / (sqrtf(cx) * sqrtf(cy));
      if (cost != cost) cost = 0.0f;
      loss += 0.4f - fminf(cost, 0.4f);
    }
    loss += M[9]; // sum relu(-Rd25)
    float pt = 0.0f;
    for (int i = 0; i < 8; ++i) {
      pt += 10.0f * relu(-dHaV[i]);
      pt += relu(-dHaJ[i]);
      pt += relu(-dHaT[i]);
      pt += relu(KELV - ToV[i]);
      pt += relu(KELV - ToJ[i]);
      pt += relu(KELV - ToT[i]);
      pt += relu(-gm[i]);
    }
    loss += pt + rowSum;
    out[0] = loss;
  }
}

extern "C" void kernel_launch(void* const* d_in, const int* in_sizes, int n_in,
                              void* d_out, int out_size, void* d_ws, size_t ws_size,
                              hipStream_t stream) {
  (void)in_sizes; (void)n_in; (void)out_size; (void)ws_size;
  const float* An  = (const float*)d_in[0];
  const float* Ac  = (const float*)d_in[1];
  const float* Aj  = (const float*)d_in[2];
  const float* Ap  = (const float*)d_in[3];
  const float* Ar  = (const float*)d_in[4];
  const float* Vc  = (const float*)d_in[5];
  const float* Jm  = (const float*)d_in[6];
  const float* Rd  = (const float*)d_in[7];
  const float* dHaV = (const float*)d_in[8];
  const float* dHaJ = (const float*)d_in[9];
  const float* dHaT = (const float*)d_in[10];
  const float* ToV  = (const float*)d_in[11];
  const float* ToJ  = (const float*)d_in[12];
  const float* ToT  = (const float*)d_in[13];
  const float* gm   = (const float*)d_in[14];
  const unsigned char* mNl = (const unsigned char*)d_in[15];
  const unsigned char* mAp = (const unsigned char*)d_in[16];
  const unsigned char* mAj = (const unsigned char*)d_in[17];

  float* ws   = (float*)d_ws;
  float* pMse = ws;                  // 1024 floats
  float* pRow = ws + 1024;           // 4096 floats
  float* pMom = ws + 1024 + 4096;    // 320 floats

  k_mse<<<1024, 256, 0, stream>>>(An, Ar, pMse);
  k_row<<<4096, 256, 0, stream>>>(Aj, Ac, Ap, mAp, mAj, pRow);
  k_mom<<<32, 256, 0, stream>>>(Vc, Jm, Rd, mNl, pMom);
  k_final<<<1, 256, 0, stream>>>(pMse, pRow, pMom, dHaV, dHaJ, dHaT,
                                 ToV, ToJ, ToT, gm, (float*)d_out);
}